// CMambaSlim_79182017069614
// MI455X (gfx1250) — compile-verified
//
#include <hip/hip_runtime.h>
#include <hip/hip_bf16.h>
#include <math.h>

// Problem constants (match reference)
#define BB     32
#define CC     64
#define LL     512
#define PPATCH 16
#define SSTR   4
#define NP     125      // (512-16)/4 + 1
#define DD     128
#define INNER  256
#define KCONV  5
#define PADK   2
#define NL     4
#define FF     96
#define EPSV   1e-5f
#define MROWS  (BB * NP)     // 4000
#define HEADK  (DD * NP)     // 16000
#define HEADN  (CC * FF)     // 6144
#define NSPLIT 16            // K-split factor for the HBM-bound head GEMM
#define KCHUNK (HEADK / NSPLIT)   // 1000

typedef float v2f __attribute__((ext_vector_type(2)));
typedef float v8f __attribute__((ext_vector_type(8)));

// ---------------------------------------------------------------------------
// fp32 WMMA: D(16x16) = A(16x4) * B(4x16) + C   (V_WMMA_F32_16X16X4_F32)
// Per-lane layout (ISA 7.12.2): m/n = lane%16, kbase = 2*(lane/16),
// A = {A[m][kbase], A[m][kbase+1]}, B = {B[kbase][n], B[kbase+1][n]},
// C/D vgpr v holds row m = v + 8*(lane/16), col n = lane%16.
// ---------------------------------------------------------------------------
__device__ __forceinline__ v8f wmma4(v2f a, v2f b, v8f c) {
  return __builtin_amdgcn_wmma_f32_16x16x4_f32(
      /*neg_a=*/false, a, /*neg_b=*/false, b,
      /*c_mod=*/(short)0, c, /*reuse_a=*/false, /*reuse_b=*/false);
}

// ---------------------------------------------------------------------------
// Patch embed: h[b,kp,d] = sum_p xp[b,kp,p]*pe_W[d,p] + pe_b[d] + sincosPE
// xp[b,kp,c*16+pp] = x[b,c,kp*4+pp]. One wave per 32x16 tile (2 M-tiles
// sharing each B load, 2 independent WMMA chains); K=1024 in steps of 4.
// ---------------------------------------------------------------------------
__global__ __launch_bounds__(32) void patch_embed_kernel(
    const float* __restrict__ x, const float* __restrict__ peW,
    const float* __restrict__ peB, float* __restrict__ h) {
  const int lane = threadIdx.x;
  const int tm   = blockIdx.x >> 3;      // 125 M-blocks of 32 rows
  const int tn   = blockIdx.x & 7;       // 8 N-tiles
  const int ml   = lane & 15;
  const int koff = (lane >> 4) << 1;     // 0 or 2

  const float* xrow[2];
#pragma unroll
  for (int mt = 0; mt < 2; ++mt) {
    int mrow = tm * 32 + mt * 16 + ml;
    int brow = mrow / NP;
    int kp   = mrow - brow * NP;
    xrow[mt] = x + (size_t)brow * CC * LL + kp * SSTR;
  }

  const int ncol = tn * 16 + ml;
  const float* wrow = peW + (size_t)ncol * (CC * PPATCH);

  v8f acc0 = {}, acc1 = {};
  for (int k4 = 0; k4 < 256; ++k4) {           // 256 * K4 = 1024
    int c  = k4 >> 2;
    int pp = ((k4 & 3) << 2) + koff;           // stays inside one channel row
    v2f bv  = *reinterpret_cast<const v2f*>(wrow + (k4 << 2) + koff);
    v2f av0 = *reinterpret_cast<const v2f*>(xrow[0] + (size_t)c * LL + pp);
    v2f av1 = *reinterpret_cast<const v2f*>(xrow[1] + (size_t)c * LL + pp);
    acc0 = wmma4(av0, bv, acc0);
    acc1 = wmma4(av1, bv, acc1);
  }

  // epilogue: bias + sincos positional encoding (branchless select)
  const float bias = peB[ncol];
  const int   j    = ncol >> 1;
  const float dv   = __expf(-9.210340371976184f * (float)j * (1.0f / 64.0f));
  const int   mb   = (lane >> 4) << 3;
#pragma unroll
  for (int mt = 0; mt < 2; ++mt) {
    const v8f acc = mt ? acc1 : acc0;
#pragma unroll
    for (int v = 0; v < 8; ++v) {
      int mg  = tm * 32 + mt * 16 + mb + v;
      int b2  = mg / NP;
      int kp2 = mg - b2 * NP;
      float ang = (float)kp2 * dv;
      float sv = __sinf(ang), cv = __cosf(ang);
      float pe = (ncol & 1) ? cv : sv;          // v_cndmask, no EXEC change
      h[(size_t)mg * DD + ncol] = acc[v] + bias + pe;
    }
  }
}

// ---------------------------------------------------------------------------
// Generic WMMA GEMM, MT M-tiles per wave sharing each B load; optional K-split
// via blockIdx.y (each y-slice accumulates K in [y*kchunk, (y+1)*kchunk) and
// writes to its own partial slab C + y*csplit_stride).
//   C[m,n] = sum_k A[m,k]*W[n,k] (+ cscale*Cin[m,n]) (+ bias[n])
// MT=2 halves weight-stream traffic and yields 2 independent WMMA chains.
// K-split exists purely for memory-level parallelism on the HBM-bound head
// GEMM (need ~18MB in flight to hit 23.3 TB/s; 384 waves alone can't).
// ---------------------------------------------------------------------------
template <int MT>
__global__ __launch_bounds__(32) void gemm_wmma_kernel(
    const float* __restrict__ A, int lda,
    const float* __restrict__ W, int ldw,
    float* __restrict__ C, int ldc, size_t csplit_stride,
    const float* __restrict__ Cin, float cscale,
    const float* __restrict__ bias,
    int kchunk, int ntn) {
  const int lane = threadIdx.x;
  const int tmb  = blockIdx.x / ntn;          // M-block of MT*16 rows
  const int tn   = blockIdx.x - tmb * ntn;
  const int ml   = lane & 15;
  const int koff = (lane >> 4) << 1;
  const int ncol = tn * 16 + ml;
  const int mb   = (lane >> 4) << 3;
  const int m0   = tmb * (MT * 16);
  const int k0   = blockIdx.y * kchunk;
  const int kend = k0 + kchunk;
  C += (size_t)blockIdx.y * csplit_stride;

  const float* arow[MT];
#pragma unroll
  for (int mt = 0; mt < MT; ++mt)
    arow[mt] = A + (size_t)(m0 + mt * 16 + ml) * lda;
  const float* wrow = W + (size_t)ncol * ldw;

  v8f acc[MT];
#pragma unroll
  for (int mt = 0; mt < MT; ++mt) acc[mt] = v8f{};
  if (Cin) {
#pragma unroll
    for (int mt = 0; mt < MT; ++mt)
#pragma unroll
      for (int v = 0; v < 8; ++v)
        acc[mt][v] = cscale * Cin[(size_t)(m0 + mt * 16 + mb + v) * ldc + ncol];
  }

  for (int k = k0; k < kend; k += 4) {
    // uniform branches: EXEC untouched (WMMA-safe); prefetch uses no LOADcnt,
    // extending the in-flight window on the streamed weight rows.
    if ((k & 63) == 0) __builtin_prefetch(wrow + k + 1024, 0, 0);
    if ((k & 255) == 0) {
#pragma unroll
      for (int mt = 0; mt < MT; ++mt)
        __builtin_prefetch(arow[mt] + k + 512, 0, 0);
    }
    v2f bv = *reinterpret_cast<const v2f*>(wrow + k + koff);
#pragma unroll
    for (int mt = 0; mt < MT; ++mt) {
      v2f av = *reinterpret_cast<const v2f*>(arow[mt] + k + koff);
      acc[mt] = wmma4(av, bv, acc[mt]);        // MT independent chains
    }
  }

  const float bv2 = bias ? bias[ncol] : 0.0f;
#pragma unroll
  for (int mt = 0; mt < MT; ++mt)
#pragma unroll
    for (int v = 0; v < 8; ++v)
      C[(size_t)(m0 + mt * 16 + mb + v) * ldc + ncol] = acc[mt][v] + bv2;
}

// ---------------------------------------------------------------------------
// Head K-split reduction: out[b,j] = sum_s part[s][b][j] + bias[j].
// ---------------------------------------------------------------------------
__global__ __launch_bounds__(256) void reduce_head_kernel(
    const float* __restrict__ part, const float* __restrict__ bias,
    float* __restrict__ out) {
  const int idx = blockIdx.x * 256 + threadIdx.x;   // BB*HEADN = 196608
  const int j   = idx % HEADN;
  float s = bias[j];
#pragma unroll
  for (int sp = 0; sp < NSPLIT; ++sp)
    s += part[(size_t)sp * (BB * HEADN) + idx];
  out[idx] = s;
}

// ---------------------------------------------------------------------------
// RMSNorm: one wave32 per row of D=128 (4 elems/lane, butterfly reduce).
// ---------------------------------------------------------------------------
__global__ __launch_bounds__(256) void rmsnorm_kernel(
    const float* __restrict__ h, const float* __restrict__ w,
    float* __restrict__ out, int nrows) {
  const int wave = blockIdx.x * (blockDim.x >> 5) + (threadIdx.x >> 5);
  if (wave >= nrows) return;
  const int lane = threadIdx.x & 31;
  const float* row = h + (size_t)wave * DD;
  float4 v = *reinterpret_cast<const float4*>(row + lane * 4);
  float ss = v.x * v.x + v.y * v.y + v.z * v.z + v.w * v.w;
#pragma unroll
  for (int off = 16; off > 0; off >>= 1) ss += __shfl_xor(ss, off, 32);
  const float inv = rsqrtf(ss * (1.0f / DD) + EPSV);
  float4 wv = *reinterpret_cast<const float4*>(w + lane * 4);
  float4 o;
  o.x = v.x * inv * wv.x; o.y = v.y * inv * wv.y;
  o.z = v.z * inv * wv.z; o.w = v.w * inv * wv.w;
  *reinterpret_cast<float4*>(out + (size_t)wave * DD + lane * 4) = o;
}

// ---------------------------------------------------------------------------
// Depthwise conv (K=5, pad 2 along NP) + SiLU. blockIdx = b*NP+kp, tid = i.
// ---------------------------------------------------------------------------
__global__ __launch_bounds__(INNER) void conv_silu_kernel(
    const float* __restrict__ a, const float* __restrict__ w,
    const float* __restrict__ cb, float* __restrict__ u) {
  const int i   = threadIdx.x;
  const int row = blockIdx.x;          // b*NP + kp
  const int b   = row / NP;
  const int kp  = row - b * NP;
  const float* w5 = w + i * KCONV;
  float acc = cb[i];
#pragma unroll
  for (int t = 0; t < KCONV; ++t) {
    int kk = kp + t - PADK;
    if (kk >= 0 && kk < NP)
      acc += a[((size_t)b * NP + kk) * INNER + i] * w5[t];
  }
  u[(size_t)row * INNER + i] = acc / (1.0f + __expf(-acc));  // SiLU
}

// ---------------------------------------------------------------------------
// Selective scan over NP (sequential) x (B*INNER parallel, coalesced) fused
// with elementwise multiply by the b-branch: sb = (gamma*s + delta*u) * b.
// ---------------------------------------------------------------------------
__global__ __launch_bounds__(INNER) void scan_kernel(
    const float* __restrict__ u, const float* __restrict__ bmat,
    const float* __restrict__ alpha, const float* __restrict__ beta,
    const float* __restrict__ gamma, const float* __restrict__ delta,
    float* __restrict__ sb) {
  const int i = threadIdx.x;           // INNER
  const int b = blockIdx.x;            // B
  const float as = 1.0f / (1.0f + __expf(-alpha[i]));
  const float be = beta[i], ga = gamma[i], de = delta[i];
  float s = 0.0f;
  const size_t base = (size_t)b * NP * INNER + i;
  for (int kp = 0; kp < NP; ++kp) {
    const size_t off = base + (size_t)kp * INNER;
    float ut = u[off];
    s = as * s + be * ut;
    sb[off] = (ga * s + de * ut) * bmat[off];
  }
}

// ---------------------------------------------------------------------------
extern "C" void kernel_launch(void* const* d_in, const int* in_sizes, int n_in,
                              void* d_out, int out_size, void* d_ws, size_t ws_size,
                              hipStream_t stream) {
  (void)in_sizes; (void)n_in; (void)out_size; (void)ws_size;
  const float* x     = (const float*)d_in[0];
  const float* peW   = (const float*)d_in[1];
  const float* peB   = (const float*)d_in[2];
  const float* normw = (const float*)d_in[3];
  const float* ipaW  = (const float*)d_in[4];
  const float* ipbW  = (const float*)d_in[5];
  const float* convW = (const float*)d_in[6];
  const float* convB = (const float*)d_in[7];
  const float* alpha = (const float*)d_in[8];
  const float* beta  = (const float*)d_in[9];
  const float* gamma = (const float*)d_in[10];
  const float* delta = (const float*)d_in[11];
  const float* opW   = (const float*)d_in[12];
  const float* normf = (const float*)d_in[13];
  const float* outW  = (const float*)d_in[14];
  const float* outB  = (const float*)d_in[15];
  float* out = (float*)d_out;

  // workspace: h(2MB) hn(2MB) a(4MB) b(4MB) u(4MB) part(12.6MB) ~= 29MB
  float* h    = (float*)d_ws;
  float* hn   = h    + (size_t)MROWS * DD;
  float* abuf = hn   + (size_t)MROWS * DD;
  float* bbuf = abuf + (size_t)MROWS * INNER;
  float* ubuf = bbuf + (size_t)MROWS * INNER;
  float* part = ubuf + (size_t)MROWS * INNER;   // NSPLIT * BB * HEADN
  float* sb   = abuf;   // reuse a-buffer after conv consumed it

  // h = patch_embed(x) + sincos PE  (125 M-blocks x 8 N-tiles)
  patch_embed_kernel<<<125 * 8, 32, 0, stream>>>(x, peW, peB, h);

  for (int l = 0; l < NL; ++l) {
    rmsnorm_kernel<<<MROWS / 8, 256, 0, stream>>>(h, normw + l * DD, hn, MROWS);
    // a = hn @ ipaW^T ; b = hn @ ipbW^T   (M=4000,K=128,N=256)
    gemm_wmma_kernel<2><<<125 * 16, 32, 0, stream>>>(
        hn, DD, ipaW + (size_t)l * INNER * DD, DD, abuf, INNER, 0,
        nullptr, 0.0f, nullptr, DD, 16);
    gemm_wmma_kernel<2><<<125 * 16, 32, 0, stream>>>(
        hn, DD, ipbW + (size_t)l * INNER * DD, DD, bbuf, INNER, 0,
        nullptr, 0.0f, nullptr, DD, 16);
    conv_silu_kernel<<<MROWS, INNER, 0, stream>>>(
        abuf, convW + (size_t)l * INNER * KCONV, convB + l * INNER, ubuf);
    scan_kernel<<<BB, INNER, 0, stream>>>(
        ubuf, bbuf, alpha + l * INNER, beta + l * INNER,
        gamma + l * INNER, delta + l * INNER, sb);
    // h = 2*h + sb @ opW^T   (reference does h = h + (h + y))
    gemm_wmma_kernel<2><<<125 * 8, 32, 0, stream>>>(
        sb, INNER, opW + (size_t)l * DD * INNER, INNER, h, DD, 0,
        h, 2.0f, nullptr, INNER, 8);
  }

  rmsnorm_kernel<<<MROWS / 8, 256, 0, stream>>>(h, normf, hn, MROWS);

  // head: out[b,j] = flat[b,:].outW[j,:] + outB[j]; flat = hn viewed (32,16000)
  // K-split x16 (grid.y) for HBM MLP: 6144 waves, out_W streamed exactly once;
  // partials reduced (+bias) below.
  gemm_wmma_kernel<2><<<dim3(384, NSPLIT), 32, 0, stream>>>(
      hn, HEADK, outW, HEADK, part, HEADN, (size_t)BB * HEADN,
      nullptr, 0.0f, nullptr, KCHUNK, 384);
  reduce_head_kernel<<<(BB * HEADN) / 256, 256, 0, stream>>>(part, outB, out);
}